// GCN_79809082294318
// MI455X (gfx1250) — compile-verified
//
#include <hip/hip_runtime.h>

// GCN 2-layer forward for MI455X (gfx1250), wave32.
//   h   = relu( Dis*(A+I)*Dis * (x@W1) + b1 )
//   out =       Dis*(A+I)*Dis * (h@W2) + b2
// GEMM1 uses V_WMMA_F32_16X16X4_F32 (full fp32 precision; GEMM is bandwidth-
// trivial so no reason to quantize). Edge aggregation is L2-resident
// (agg buffers = 51MB << 192MB L2) scatter-add with f32 atomics.

typedef __attribute__((ext_vector_type(2))) float v2f;
typedef __attribute__((ext_vector_type(8))) float v8f;

static constexpr int KF = 128;   // input features
static constexpr int KH = 128;   // hidden features

// ---------------------------------------------------------------- degrees ---
__global__ void k_init_deg(float* __restrict__ deg, int N) {
  int i = blockIdx.x * blockDim.x + threadIdx.x;
  if (i < N) deg[i] = 1.0f;               // self-loop contributes 1
}

__global__ void k_count_deg(const int* __restrict__ dst, float* __restrict__ deg, int E) {
  int e = blockIdx.x * blockDim.x + threadIdx.x;
  if (e < E) atomicAdd(&deg[dst[e]], 1.0f);
}

__global__ void k_rsqrt(float* __restrict__ deg, int N) {
  int i = blockIdx.x * blockDim.x + threadIdx.x;
  if (i < N) deg[i] = rsqrtf(fmaxf(deg[i], 1.0f));   // deg -> dis in place
}

// ------------------------------------------------- GEMM1 via fp32 WMMA ------
// One block = 16 rows x 128 cols. 8 waves, each wave owns a 16x16 tile.
// W1 (128x128 f32 = 64KB) staged into LDS once per block.
// Epilogue writes hn = (x@W1)*dis[row] and agg1 = hn*dis[row] + b1
// (self-loop term + bias folded into the aggregation initializer).
__global__ void __launch_bounds__(256)
k_gemm_wmma(const float* __restrict__ x, const float* __restrict__ W1,
            const float* __restrict__ b1, const float* __restrict__ dis,
            float* __restrict__ hn, float* __restrict__ agg1, int N) {
  __shared__ float sW[KF * KH];           // 64 KB of the 320 KB/WGP LDS
  {
    const float4* Wv = (const float4*)W1;
    float4* sWv = (float4*)sW;
    const int tid = threadIdx.x;
#pragma unroll
    for (int i = 0; i < (KF * KH / 4) / 256; ++i)   // 16 iters of float4
      sWv[i * 256 + tid] = Wv[i * 256 + tid];
  }
  __syncthreads();

  const int tid  = threadIdx.x;
  const int wave = tid >> 5;
  const int lane = tid & 31;
  const int hh   = lane >> 4;        // lane half (0 or 1)
  const int l15  = lane & 15;
  const int c0   = wave << 4;        // this wave's 16-column tile
  const int r0   = (int)blockIdx.x << 4;

  int rowA = r0 + l15;               // A-matrix: M = lane&15
  if (rowA >= N) rowA = N - 1;       // clamp loads; WMMA needs EXEC all-1s
  const float* xrow = x + (size_t)rowA * KF;

  v8f acc = {};
#pragma unroll
  for (int k0 = 0; k0 < KF; k0 += 4) {
    const int ka = k0 + 2 * hh;      // A/B VGPR layout: K = vgpr + 2*half
    v2f a;
    a[0] = xrow[ka + 0];
    a[1] = xrow[ka + 1];
    v2f b;
    b[0] = sW[(ka + 0) * KH + c0 + l15];
    b[1] = sW[(ka + 1) * KH + c0 + l15];
    acc = __builtin_amdgcn_wmma_f32_16x16x4_f32(
        /*neg_a=*/false, a, /*neg_b=*/false, b,
        /*c_mod=*/(short)0, acc, /*reuse_a=*/false, /*reuse_b=*/false);
  }

  const float bias = b1[c0 + l15];
#pragma unroll
  for (int v = 0; v < 8; ++v) {      // C/D layout: M = vgpr + 8*half, N = lane&15
    const int rr = r0 + v + 8 * hh;
    if (rr < N) {
      const float di   = dis[rr];
      const float hval = acc[v] * di;                  // (x@W1)*dis[src]
      hn  [(size_t)rr * KH + c0 + l15] = hval;
      agg1[(size_t)rr * KH + c0 + l15] = hval * di + bias;  // self-loop + b1
    }
  }
}

// -------------------------------------------- edge scatter, 128 features ----
// One wave per edge: lane L handles features [4L, 4L+4). Gathers a full
// 512B row of hn (coalesced) and scatter-adds into agg1[dst].
__global__ void k_scatter_feat(const float* __restrict__ hn,
                               const int* __restrict__ src,
                               const int* __restrict__ dst,
                               const float* __restrict__ dis,
                               float* __restrict__ agg1, int E) {
  const long long gid = (long long)blockIdx.x * blockDim.x + threadIdx.x;
  const int e    = (int)(gid >> 5);
  const int lane = (int)(gid & 31);
  if (e >= E) return;
  const int s = src[e];
  const int d = dst[e];
  const float nd = dis[d];           // hn already carries dis[src]
  const float4 h = ((const float4*)(hn + (size_t)s * KH))[lane];
  float* o = agg1 + (size_t)d * KH + lane * 4;
  atomicAdd(o + 0, h.x * nd);
  atomicAdd(o + 1, h.y * nd);
  atomicAdd(o + 2, h.z * nd);
  atomicAdd(o + 3, h.w * nd);
}

// --------------------------- layer 2: relu + 128->1 matvec + out init -------
// One wave per node; wave-level shuffle reduction (wave32).
__global__ void k_node_layer2(const float* __restrict__ agg1,
                              const float* __restrict__ W2,
                              const float* __restrict__ b2,
                              const float* __restrict__ dis,
                              float* __restrict__ zn, float* __restrict__ out, int N) {
  const long long gid = (long long)blockIdx.x * blockDim.x + threadIdx.x;
  const int i    = (int)(gid >> 5);
  const int lane = (int)(gid & 31);
  if (i >= N) return;
  const float4 a = ((const float4*)(agg1 + (size_t)i * KH))[lane];  // b1 already folded in
  const float4 w = ((const float4*)W2)[lane];
  float s = fmaxf(a.x, 0.f) * w.x + fmaxf(a.y, 0.f) * w.y +
            fmaxf(a.z, 0.f) * w.z + fmaxf(a.w, 0.f) * w.w;
#pragma unroll
  for (int off = 16; off > 0; off >>= 1) s += __shfl_xor(s, off, 32);
  if (lane == 0) {
    const float di = dis[i];
    zn[i]  = s * di;                 // z*dis[src], consumed by edge scatter
    out[i] = s * di * di + b2[0];    // self-loop + bias
  }
}

// ------------------------------------------- edge scatter, scalar layer -----
__global__ void k_scatter_scalar(const float* __restrict__ zn,
                                 const int* __restrict__ src,
                                 const int* __restrict__ dst,
                                 const float* __restrict__ dis,
                                 float* __restrict__ out, int E) {
  int e = blockIdx.x * blockDim.x + threadIdx.x;
  if (e >= E) return;
  const int d = dst[e];
  atomicAdd(&out[d], zn[src[e]] * dis[d]);
}

// ---------------------------------------------------------------------------
extern "C" void kernel_launch(void* const* d_in, const int* in_sizes, int n_in,
                              void* d_out, int out_size, void* d_ws, size_t ws_size,
                              hipStream_t stream) {
  (void)n_in; (void)out_size; (void)ws_size;
  const float* x  = (const float*)d_in[0];   // [N,128]
  const int*   ei = (const int*)d_in[1];     // [2,E] row-major
  const float* W1 = (const float*)d_in[2];   // [128,128]
  const float* b1 = (const float*)d_in[3];   // [128]
  const float* W2 = (const float*)d_in[4];   // [128,1]
  const float* b2 = (const float*)d_in[5];   // [1]

  const int N = in_sizes[0] / KF;
  const int E = in_sizes[1] / 2;
  const int* srcI = ei;
  const int* dstI = ei + E;

  // workspace: dis[N] | zn[N] | hn[N*128] | agg1[N*128]  (~103 MB)
  float* ws   = (float*)d_ws;
  float* dis  = ws;
  float* zn   = ws + N;
  float* hn   = ws + 2 * (size_t)N;
  float* agg1 = hn + (size_t)N * KH;
  float* out  = (float*)d_out;

  const int T = 256;
  k_init_deg <<<(N + T - 1) / T, T, 0, stream>>>(dis, N);
  k_count_deg<<<(E + T - 1) / T, T, 0, stream>>>(dstI, dis, E);
  k_rsqrt    <<<(N + T - 1) / T, T, 0, stream>>>(dis, N);

  k_gemm_wmma<<<(N + 15) / 16, 256, 0, stream>>>(x, W1, b1, dis, hn, agg1, N);

  const long long sfT = (long long)E * 32;
  k_scatter_feat<<<(unsigned)((sfT + T - 1) / T), T, 0, stream>>>(hn, srcI, dstI, dis, agg1, E);

  const long long nT = (long long)N * 32;
  k_node_layer2<<<(unsigned)((nT + T - 1) / T), T, 0, stream>>>(agg1, W2, b2, dis, zn, out, N);

  k_scatter_scalar<<<(E + T - 1) / T, T, 0, stream>>>(zn, srcI, dstI, dis, out, E);
}